// CRFNLLHead_6932077216416
// MI455X (gfx1250) — compile-verified
//
#include <hip/hip_runtime.h>
#include <math.h>

#define Bv 512
#define Tv 1024
#define Nv 48

typedef __attribute__((ext_vector_type(2))) float v2f;
typedef __attribute__((ext_vector_type(8))) float v8f;
typedef __attribute__((ext_vector_type(4))) unsigned int v4u;
typedef __attribute__((ext_vector_type(4))) int v4i;
typedef __attribute__((ext_vector_type(8))) int v8i;

#if __has_builtin(__builtin_amdgcn_tensor_load_to_lds) && \
    __has_builtin(__builtin_amdgcn_s_wait_tensorcnt)
#define USE_TDM 1
#else
#define USE_TDM 0
#endif

// ---------------------------------------------------------------------------
// Kernel 1: gold-path score  seq[b] = sum_t unary + sum_t binary   (masked)
// ---------------------------------------------------------------------------
__global__ __launch_bounds__(128) void crf_seq_score(
    const float* __restrict__ pot, const int* __restrict__ y,
    const int* __restrict__ len, const float* __restrict__ trans,
    float* __restrict__ seq) {
  __shared__ float red[128];
  const int b = blockIdx.x;
  const int tid = threadIdx.x;
  const int L = len[b];
  float acc = 0.f;
  for (int t = tid; t < Tv; t += 128) {
    if (t < L) {
      const int tag = y[b * Tv + t];
      acc += pot[((long)b * Tv + t) * Nv + tag];
      if (t >= 1) {
        const int ptag = y[b * Tv + t - 1];
        acc += trans[ptag * Nv + tag];
      }
    }
  }
  red[tid] = acc;
  __syncthreads();
  for (int s = 64; s > 0; s >>= 1) {
    if (tid < s) red[tid] += red[tid + s];
    __syncthreads();
  }
  if (tid == 0) seq[b] = red[0];
}

// ---------------------------------------------------------------------------
// Kernel 2: forward algorithm, exp-domain WMMA matmul + TDM double buffering.
// One wave (32 threads) handles 16 batch rows; grid = B/16 = 32 blocks.
//   alpha'_{b,j} = m_b + log( sum_i exp(alpha_{b,i}-m_b) * exp(trans_{i,j}) )
//                + pot[b,t,j]                                   (if t < len_b)
// While step t computes (36x V_WMMA_F32_16X16X4_F32 + exp/log), the Tensor
// Data Mover streams the 16x48 unary tile for step t+1 into the other LDS
// buffer (tracked with TENSORcnt), hiding the HBM round trip of the only
// fresh data each step needs.
// ---------------------------------------------------------------------------
__global__ __launch_bounds__(32) void crf_forward(
    const float* __restrict__ pot, const int* __restrict__ len,
    const float* __restrict__ trans, const float* __restrict__ seq,
    float* __restrict__ out) {
  __shared__ float alpha[16][Nv];      // 16 batches x 48 states
  __shared__ float ptile[2][16][Nv];   // double-buffered unary tiles
  __shared__ float mrow[16];           // per-batch running max
  __shared__ float pmax[32];           // half-reduction scratch

  const int lane = threadIdx.x;  // 0..31 (wave32)
  const int lo = lane & 15;
  const int hi = lane >> 4;  // 0 or 1
  const int b0 = blockIdx.x * 16;

  // --- tile fetch for timestep t1 into ptile[buf]  (TDM or manual fallback)
#if USE_TDM
  auto load_tile = [&](int t1, int buf) {
    const unsigned long long ga =
        (unsigned long long)(const void*)(pot + ((long)b0 * Tv + t1) * Nv);
    const unsigned lds_off = (unsigned)(uintptr_t)(&ptile[buf][0][0]);
    v4u g0;
    g0.x = 1u;                              // count=1 valid user descriptor
    g0.y = lds_off;                         // lds_addr (bytes)
    g0.z = (unsigned)ga;                    // global_addr[31:0]
    g0.w = (unsigned)((ga >> 32) & 0x01ffffffull) | (2u << 30);  // type=2
    v8i g1;
    g1[0] = (int)(2u << 16);   // workgroup_mask=0, data_size=2 (4 bytes)
    g1[1] = (int)(48u << 16);  // tensor_dim0 = 48      (bits 79:48)
    g1[2] = (int)(16u << 16);  // tensor_dim1 = 16      (bits 111:80)
    g1[3] = (int)(48u << 16);  // tile_dim0   = 48      (bits 127:112)
    g1[4] = 16;                // tile_dim1   = 16      (bits 143:128)
    g1[5] = Tv * Nv;           // tensor_dim0_stride = 49152 (bits 207:160)
    g1[6] = 0;
    g1[7] = 0;
    v4i gz = {0, 0, 0, 0};     // groups 2/3 unused (2D tensor)
#if defined(__clang_major__) && __clang_major__ >= 23
    v8i gz8 = {0, 0, 0, 0, 0, 0, 0, 0};
    __builtin_amdgcn_tensor_load_to_lds(g0, g1, gz, gz, gz8, 0);
#else
    __builtin_amdgcn_tensor_load_to_lds(g0, g1, gz, gz, 0);
#endif
  };
#else
  auto load_tile = [&](int t1, int buf) {
#pragma unroll
    for (int i = 0; i < 24; ++i) {
      const int idx = lane + 32 * i;
      const int bb = idx / Nv, j = idx % Nv;
      ptile[buf][bb][j] = pot[((long)(b0 + bb) * Tv + t1) * Nv + j];
    }
  };
#endif

  // --- Loop-invariant B fragments: exp(transitions) in WMMA 4x16 layout.
  // VGPR0 holds row K=0 (lanes 0-15) / K=2 (lanes 16-31); VGPR1 K=1 / K=3.
  v2f bfrag[3][12];
#pragma unroll
  for (int nt = 0; nt < 3; ++nt) {
#pragma unroll
    for (int kk = 0; kk < 12; ++kk) {
      const int k0 = 4 * kk + 2 * hi;
      const int n = nt * 16 + lo;
      bfrag[nt][kk].x = __expf(trans[k0 * Nv + n]);
      bfrag[nt][kk].y = __expf(trans[(k0 + 1) * Nv + n]);
    }
  }

  // --- Per-lane row lengths for the 8 batch rows this lane masks (C layout).
  int Lr[8];
#pragma unroll
  for (int r = 0; r < 8; ++r) Lr[r] = len[b0 + r + 8 * hi];

  // --- alpha0 = potentials[:, 0, :]; kick off pipeline with tile for t=1.
  load_tile(1, 1);
  for (int idx = lane; idx < 16 * Nv; idx += 32) {
    const int bb = idx / Nv, j = idx % Nv;
    alpha[bb][j] = pot[((long)(b0 + bb) * Tv) * Nv + j];
  }
  __syncthreads();

  for (int t = 1; t < Tv; ++t) {
    const int cur = t & 1;
#if USE_TDM
    __builtin_amdgcn_s_wait_tensorcnt(0);  // tile t resident in LDS
#endif
    __syncthreads();
    if (t + 1 < Tv) load_tile(t + 1, cur ^ 1);  // overlap with this step

    // ---- per-batch max of alpha (stability shift)
    float pm = -INFINITY;
#pragma unroll
    for (int jj = 0; jj < 24; ++jj) pm = fmaxf(pm, alpha[lo][hi * 24 + jj]);
    pmax[lane] = pm;
    __syncthreads();
    if (lane < 16) mrow[lane] = fmaxf(pmax[lane], pmax[lane + 16]);
    __syncthreads();

    // ---- exp-domain matmul: D[16x48] = exp(alpha-m)[16x48] @ exp(T)[48x48]
    v8f acc0 = {}, acc1 = {}, acc2 = {};
    const float mme = mrow[lo];
#pragma unroll
    for (int kk = 0; kk < 12; ++kk) {
      v2f a;
      const int k0 = 4 * kk + 2 * hi;  // A 16x4 layout: K=0/1 | K=2/3
      a.x = __expf(alpha[lo][k0] - mme);
      a.y = __expf(alpha[lo][k0 + 1] - mme);
      acc0 = __builtin_amdgcn_wmma_f32_16x16x4_f32(false, a, false,
                                                   bfrag[0][kk], (short)0,
                                                   acc0, false, false);
      acc1 = __builtin_amdgcn_wmma_f32_16x16x4_f32(false, a, false,
                                                   bfrag[1][kk], (short)0,
                                                   acc1, false, false);
      acc2 = __builtin_amdgcn_wmma_f32_16x16x4_f32(false, a, false,
                                                   bfrag[2][kk], (short)0,
                                                   acc2, false, false);
    }
    __syncthreads();  // all alpha reads complete before overwrite

    // ---- back to log-domain + unary (from LDS tile), masked write-back.
    // C/D layout: vgpr r, lane -> (M = r + 8*hi, N = lo) within each N-tile.
#pragma unroll
    for (int r = 0; r < 8; ++r) {
      const int b16 = r + 8 * hi;
      if (t < Lr[r]) {
        const float mm = mrow[b16];
        alpha[b16][lo] = mm + __logf(acc0[r]) + ptile[cur][b16][lo];
        alpha[b16][16 + lo] = mm + __logf(acc1[r]) + ptile[cur][b16][16 + lo];
        alpha[b16][32 + lo] = mm + __logf(acc2[r]) + ptile[cur][b16][32 + lo];
      }
    }
    __syncthreads();
  }

  // ---- log_norm[b] = logsumexp_j alpha[b, j];  out = log_norm - seq_score
  float pm = -INFINITY;
#pragma unroll
  for (int jj = 0; jj < 24; ++jj) pm = fmaxf(pm, alpha[lo][hi * 24 + jj]);
  pmax[lane] = pm;
  __syncthreads();
  if (lane < 16) mrow[lane] = fmaxf(pmax[lane], pmax[lane + 16]);
  __syncthreads();
  float s = 0.f;
#pragma unroll
  for (int jj = 0; jj < 24; ++jj)
    s += __expf(alpha[lo][hi * 24 + jj] - mrow[lo]);
  pmax[lane] = s;
  __syncthreads();
  if (lane < 16) {
    const float log_norm = mrow[lane] + __logf(pmax[lane] + pmax[lane + 16]);
    out[b0 + lane] = log_norm - seq[b0 + lane];
  }
}

// ---------------------------------------------------------------------------
extern "C" void kernel_launch(void* const* d_in, const int* in_sizes, int n_in,
                              void* d_out, int out_size, void* d_ws,
                              size_t ws_size, hipStream_t stream) {
  const float* pot = (const float*)d_in[0];   // [B, T, N] f32
  const int* y = (const int*)d_in[1];         // [B, T] i32
  const int* len = (const int*)d_in[2];       // [B] i32
  const float* trans = (const float*)d_in[3]; // [N, N] f32
  float* out = (float*)d_out;                 // [B] f32
  float* seq = (float*)d_ws;                  // [B] f32 scratch

  crf_seq_score<<<Bv, 128, 0, stream>>>(pot, y, len, trans, seq);
  crf_forward<<<Bv / 16, 32, 0, stream>>>(pot, len, trans, seq, out);
}